// SphericalVectorPool_80023830659728
// MI455X (gfx1250) — compile-verified
//
#include <hip/hip_runtime.h>
#include <hip/hip_bf16.h>

typedef __attribute__((ext_vector_type(2))) float v2f;
typedef __attribute__((ext_vector_type(8))) float v8f;

#define NSCALE 16
#define ATILE  16
#define ICHUNK 4
#define IRANGE 128   // i-points handled per block

__global__ void svp_zero_kernel(float* __restrict__ out, int n) {
    int i = blockIdx.x * blockDim.x + threadIdx.x;
    if (i < n) out[i] = 0.0f;
}

__launch_bounds__(32)
__global__ void svp_main_kernel(const float* __restrict__ f,
                                const float* __restrict__ coords,
                                const float* __restrict__ out_coords,
                                const float* __restrict__ mu,
                                const float* __restrict__ r_norms,
                                const float* __restrict__ a_norms_0,
                                const float* __restrict__ a_norms_1,
                                float* __restrict__ out) {
    __shared__ float cf[IRANGE][4];               // {cx, cy, cz, f}
    __shared__ float ft[IRANGE][4];               // {f*cx, f*cy, f*cz, f}
    __shared__ float dpack[NSCALE][ATILE][4];     // unpacked D columns 0..3

    const int lane  = (int)threadIdx.x;   // 0..31, one wave per block
    const int half  = lane >> 4;          // 0 or 1 (half-wave)
    const int lrow  = lane & 15;          // row a-index / column n-index
    const int aBase = blockIdx.x * ATILE;
    const int iBase = blockIdx.y * IRANGE;

    // ---- preload source points into LDS ----
    for (int i = lane; i < IRANGE; i += 32) {
        float cx = coords[(iBase + i) * 3 + 0];
        float cy = coords[(iBase + i) * 3 + 1];
        float cz = coords[(iBase + i) * 3 + 2];
        float fv = f[iBase + i];
        cf[i][0] = cx;      cf[i][1] = cy;      cf[i][2] = cz;      cf[i][3] = fv;
        ft[i][0] = fv * cx; ft[i][1] = fv * cy; ft[i][2] = fv * cz; ft[i][3] = fv;
    }
    __syncthreads();

    // per-lane output coordinate (row a = aBase + lrow; both halves same a)
    const int   a  = aBase + lrow;
    const float Rx = out_coords[a * 3 + 0];
    const float Ry = out_coords[a * 3 + 1];
    const float Rz = out_coords[a * 3 + 2];

    // uniform per-scale exp2 coefficients: exp(-mu*r) = exp2(r * (-mu*log2 e))
    float muln[NSCALE];
#pragma unroll
    for (int s = 0; s < NSCALE; ++s)
        muln[s] = -1.44269504088896340736f * mu[s];

    v8f   acc[NSCALE];   // D accumulators: cols 0..2 = P_xyz, col 3 = Q
    float acc0[NSCALE];  // l=0 accumulator (per-lane partial, reduced across halves)
#pragma unroll
    for (int s = 0; s < NSCALE; ++s) { acc[s] = (v8f)0.0f; acc0[s] = 0.0f; }

    for (int ic = 0; ic < IRANGE; ic += ICHUNK) {
        // This lane covers K slots {2*half, 2*half+1} of the chunk
        const int i0 = ic + 2 * half;
        const int i1 = i0 + 1;
        float c0x = cf[i0][0], c0y = cf[i0][1], c0z = cf[i0][2], f0 = cf[i0][3];
        float c1x = cf[i1][0], c1y = cf[i1][1], c1z = cf[i1][2], f1 = cf[i1][3];

        float d0x = c0x - Rx, d0y = c0y - Ry, d0z = c0z - Rz;
        float d1x = c1x - Rx, d1y = c1y - Ry, d1z = c1z - Rz;
        float r20 = d0x * d0x + d0y * d0y + d0z * d0z;
        float r21 = d1x * d1x + d1y * d1y + d1z * d1z;
        float ir0 = __builtin_amdgcn_rsqf(r20);   // 1/r (also used for H)
        float ir1 = __builtin_amdgcn_rsqf(r21);
        float r0  = r20 * ir0;                    // r
        float r1  = r21 * ir1;

        // B matrix (4x16 f32, 2 VGPRs): V0 = rows {k0 | k2}, V1 = rows {k1 | k3}
        // column n = lrow: features {f*cx, f*cy, f*cz, f} for n<4, else 0
        const int nIdx = lrow & 3;
        float b0 = (lrow < 4) ? ft[i0][nIdx] : 0.0f;
        float b1 = (lrow < 4) ? ft[i1][nIdx] : 0.0f;
        v2f B; B.x = b0; B.y = b1;

#pragma unroll
        for (int s = 0; s < NSCALE; ++s) {
            float e0 = __builtin_amdgcn_exp2f(r0 * muln[s]);   // exp(-mu_s * r)
            float e1 = __builtin_amdgcn_exp2f(r1 * muln[s]);
            // l=0 partial: sum_i exp(-mu r) * f_i (this lane's K slots)
            acc0[s] += e0 * f0 + e1 * f1;
            // A matrix (16x4 f32): lane m -> A[m, 2*half + {0,1}] = H = e/r
            v2f Am; Am.x = e0 * ir0; Am.y = e1 * ir1;
            acc[s] = __builtin_amdgcn_wmma_f32_16x16x4_f32(
                false, Am, false, B, (short)0, acc[s], false, false);
        }
    }

    // ---- cross-half reduce l=0 partials (lane m + lane m+16) ----
    float acc0red[NSCALE];
#pragma unroll
    for (int s = 0; s < NSCALE; ++s)
        acc0red[s] = acc0[s] + __shfl_xor(acc0[s], 16, 32);

    // ---- spill D columns 0..3 to LDS: D(m,n) at lane n+16*(m/8), vgpr m%8 ----
#pragma unroll
    for (int s = 0; s < NSCALE; ++s) {
        if (lrow < 4) {
#pragma unroll
            for (int v = 0; v < 8; ++v) {
                int m = v + 8 * half;
                dpack[s][m][lrow] = acc[s][v];
            }
        }
    }
    __syncthreads();

    // ---- combine + scale + atomic accumulate into global output ----
    if (half == 0) {
        const float a0v = a_norms_0[0];
#pragma unroll
        for (int s = 0; s < NSCALE; ++s) {
            float rn = r_norms[s];
            // out0[a, s]
            atomicAdd(&out[a * (4 * NSCALE) + s], acc0red[s] * rn * a0v);
            // out1[a, s, d] = (P_d - R_d * Q) * rn * a1_d
            float Px = dpack[s][lrow][0];
            float Py = dpack[s][lrow][1];
            float Pz = dpack[s][lrow][2];
            float Q  = dpack[s][lrow][3];
            float base = (float)(a * (4 * NSCALE) + NSCALE + s * 3);
            (void)base;
            atomicAdd(&out[a * (4 * NSCALE) + NSCALE + s * 3 + 0],
                      (Px - Rx * Q) * rn * a_norms_1[0]);
            atomicAdd(&out[a * (4 * NSCALE) + NSCALE + s * 3 + 1],
                      (Py - Ry * Q) * rn * a_norms_1[1]);
            atomicAdd(&out[a * (4 * NSCALE) + NSCALE + s * 3 + 2],
                      (Pz - Rz * Q) * rn * a_norms_1[2]);
        }
    }
}

extern "C" void kernel_launch(void* const* d_in, const int* in_sizes, int n_in,
                              void* d_out, int out_size, void* d_ws, size_t ws_size,
                              hipStream_t stream) {
    const float* f          = (const float*)d_in[0];
    const float* coords     = (const float*)d_in[1];
    const float* out_coords = (const float*)d_in[2];
    const float* mu         = (const float*)d_in[3];
    const float* r_norms    = (const float*)d_in[4];
    const float* a_norms_0  = (const float*)d_in[5];
    const float* a_norms_1  = (const float*)d_in[6];
    float* out = (float*)d_out;

    const int N = in_sizes[0];       // B*N = 2048
    const int A = in_sizes[2] / 3;   // 2048

    // zero output (atomic accumulation of i-split partials follows)
    svp_zero_kernel<<<(out_size + 255) / 256, 256, 0, stream>>>(out, out_size);

    dim3 grid(A / ATILE, N / IRANGE);   // 128 a-tiles x 16 i-splits
    svp_main_kernel<<<grid, 32, 0, stream>>>(f, coords, out_coords, mu, r_norms,
                                             a_norms_0, a_norms_1, out);
}